// DiscreteContinuousDecoder_6614249636090
// MI455X (gfx1250) — compile-verified
//
#include <hip/hip_runtime.h>

typedef __attribute__((ext_vector_type(2))) float v2f;
typedef __attribute__((ext_vector_type(4))) float v4f;
typedef __attribute__((ext_vector_type(8))) float v8f;

#define NLAT_IN  480
#define NLON_IN  960
#define NLAT_OUT 721
#define NLON_OUT 1440
#define KSIZE    9
#define NNZ      20
#define C_IN     32
#define C_OUT    32
#define OUT_CS   (NLAT_OUT * NLON_OUT)  // output channel stride (channel-first out)
#define NT       (NLON_OUT / 16)        // 90 column tiles per row
#define NTW      6                      // tiles per compute wave (15 waves * 6 = 90)

// ---------------------------------------------------------------------------
// Kernel 1: ResampleS2, channel-LAST staging layout:
//   xr[(ho*1440 + wo)*32 + c]
// One thread per (ho,wo) pixel handles all 32 channels; writes 8 x b128
// contiguous (a full 128-byte block per lane).
// ---------------------------------------------------------------------------
__global__ void resample_kernel(const float* __restrict__ x, float* __restrict__ xr) {
    int idx = blockIdx.x * blockDim.x + threadIdx.x;
    const int total = NLAT_OUT * NLON_OUT;
    if (idx >= total) return;

    int wo = idx % NLON_OUT;
    int ho = idx / NLON_OUT;

    // pos_h = linspace(0, 479, 721)[ho]
    float ph = (float)ho * ((float)(NLAT_IN - 1) / (float)(NLAT_OUT - 1));
    int h0 = (int)floorf(ph);
    if (h0 > NLAT_IN - 2) h0 = NLAT_IN - 2;
    if (h0 < 0) h0 = 0;
    float fh = ph - (float)h0;
    float gh = 1.0f - fh;

    // pos_w = wo * (960/1440)
    float pw = (float)wo * ((float)NLON_IN / (float)NLON_OUT);
    int w0 = (int)floorf(pw);
    float fw = pw - (float)w0;
    float gw = 1.0f - fw;
    w0 = w0 % NLON_IN;
    int w1 = (w0 + 1) % NLON_IN;

    const int r0 = h0 * NLON_IN;
    const int r1 = r0 + NLON_IN;
    float* dst = xr + (size_t)idx * C_IN;

#pragma unroll
    for (int c4 = 0; c4 < C_IN / 4; ++c4) {
        v4f v;
#pragma unroll
        for (int j = 0; j < 4; ++j) {
            const float* xc = x + (size_t)(c4 * 4 + j) * (NLAT_IN * NLON_IN);
            float a = xc[r0 + w0] * gh + xc[r1 + w0] * fh;
            float b = xc[r0 + w1] * gh + xc[r1 + w1] * fh;
            v[j] = a * gw + b * fw;
        }
        *(v4f*)(dst + c4 * 4) = v;
    }
}

// ---------------------------------------------------------------------------
// Kernel 2: DISCO conv via V_WMMA_F32_16X16X4_F32.
// One 512-thread block (16 waves) per output row ho.
//  Phase 1: weff into LDS as paired layout [nz][cpair][o]{c_even,c_odd} (80 KB)
//           -> each A fragment is a single ds_load_b64.
//  Phase 2: 15 compute waves, each owns 6 register-resident 32x16 output
//           tiles (96 acc VGPRs). Loop nz (20) -> ks (8): 2 ds_load_b64 (A),
//           6 global_load_b64 (B), 12 WMMAs. Nothing is outer-loop invariant,
//           so no spills; all control flow around WMMAs is wave-uniform.
// ---------------------------------------------------------------------------
__launch_bounds__(512, 1)
__global__ void disco_kernel(const float* __restrict__ xr,
                             const float* __restrict__ weight,
                             const float* __restrict__ psi_vals,
                             const int*   __restrict__ psi_hi,
                             const int*   __restrict__ psi_dw,
                             float* __restrict__ out) {
    __shared__ float s_A[NNZ * 16 * C_OUT * 2];  // [nz][cpair][o][2] : 80 KB
    __shared__ float s_psi[KSIZE * NNZ];
    __shared__ int   s_hi[NNZ];
    __shared__ int   s_dw[NNZ];

    const int ho  = blockIdx.x;
    const int tid = threadIdx.x;

    if (tid < KSIZE * NNZ) {
        int k  = tid / NNZ;
        int nz = tid % NNZ;
        s_psi[tid] = psi_vals[k * (NLAT_OUT * NNZ) + ho * NNZ + nz];
    }
    if (tid < NNZ) {
        s_hi[tid] = psi_hi[ho * NNZ + tid];
        s_dw[tid] = psi_dw[ho * NNZ + tid];
    }
    __syncthreads();

    // weff[o,c] = sum_k weight[o][c][k] * psi[k][nz]; 1024 (o,c) pairs / 512 thr
    for (int p = tid; p < C_OUT * C_IN; p += 512) {
        int o = p & 31;
        int c = p >> 5;
        float w[KSIZE];
#pragma unroll
        for (int k = 0; k < KSIZE; ++k)
            w[k] = weight[o * (C_IN * KSIZE) + c * KSIZE + k];
#pragma unroll 4
        for (int nz = 0; nz < NNZ; ++nz) {
            float v = 0.0f;
#pragma unroll
            for (int k = 0; k < KSIZE; ++k)
                v += w[k] * s_psi[k * NNZ + nz];
            s_A[(((nz * 16 + (c >> 1)) * C_OUT) + o) * 2 + (c & 1)] = v;
        }
    }
    __syncthreads();

    const int wave = tid >> 5;
    const int lane = tid & 31;
    const int mh   = lane >> 4;   // fragment half: K/M split across lane groups
    const int nloc = lane & 15;

    if (wave >= 15) return;       // 15 waves * 6 tiles = 90 tiles, no guards

    const v2f* s_Av = (const v2f*)s_A;

    v8f acc[NTW][2];
#pragma unroll
    for (int i = 0; i < NTW; ++i) {
        acc[i][0] = (v8f){0.f, 0.f, 0.f, 0.f, 0.f, 0.f, 0.f, 0.f};
        acc[i][1] = (v8f){0.f, 0.f, 0.f, 0.f, 0.f, 0.f, 0.f, 0.f};
    }

#pragma unroll 1
    for (int nz = 0; nz < NNZ; ++nz) {
        const int hi = s_hi[nz];
        const int dw = s_dw[nz];

        // Per-tile base pointers for this nz (channel-last layout)
        const float* bb[NTW];
#pragma unroll
        for (int i = 0; i < NTW; ++i) {
            const int t = wave + 15 * i;
            int col = dw + t * 16 + nloc;             // < 2*NLON_OUT
            if (col >= NLON_OUT) col -= NLON_OUT;     // periodic shift
            bb[i] = xr + ((size_t)hi * NLON_OUT + col) * C_IN + 2 * mh;
        }

        const int abase = (nz * 16 + mh) * C_OUT;     // v2f index, ks = 0
#pragma unroll
        for (int ks = 0; ks < 8; ++ks) {
            // A (16x4): lane -> M = nloc (+16 for acc1), K = vgpr + 2*mh
            v2f a0 = s_Av[abase + 2 * ks * C_OUT + nloc];
            v2f a1 = s_Av[abase + 2 * ks * C_OUT + 16 + nloc];
#pragma unroll
            for (int i = 0; i < NTW; ++i) {
                // B (4x16): lane -> N = nloc, K = vgpr + 2*mh; one b64 load
                v2f b = *(const v2f*)(bb[i] + 4 * ks);
                acc[i][0] = __builtin_amdgcn_wmma_f32_16x16x4_f32(
                    false, a0, false, b, (short)0, acc[i][0], false, false);
                acc[i][1] = __builtin_amdgcn_wmma_f32_16x16x4_f32(
                    false, a1, false, b, (short)0, acc[i][1], false, false);
            }
        }
    }

    // D (16x16): vgpr v, lane -> M = v + 8*mh, N = nloc. Output is channel-first.
#pragma unroll
    for (int i = 0; i < NTW; ++i) {
        const int t = wave + 15 * i;
        float* dst = out + ho * NLON_OUT + t * 16 + nloc;
#pragma unroll
        for (int v = 0; v < 8; ++v) {
            dst[(size_t)(v + 8 * mh) * OUT_CS]      = acc[i][0][v];
            dst[(size_t)(16 + v + 8 * mh) * OUT_CS] = acc[i][1][v];
        }
    }
}

// ---------------------------------------------------------------------------
extern "C" void kernel_launch(void* const* d_in, const int* in_sizes, int n_in,
                              void* d_out, int out_size, void* d_ws, size_t ws_size,
                              hipStream_t stream) {
    const float* x        = (const float*)d_in[0];
    const float* weight   = (const float*)d_in[1];
    const float* psi_vals = (const float*)d_in[2];
    const int*   psi_hi   = (const int*)d_in[3];
    const int*   psi_dw   = (const int*)d_in[4];
    float* out = (float*)d_out;
    float* xr  = (float*)d_ws;  // 32*721*1440*4 = 132.9 MB scratch, channel-last

    const int npix = NLAT_OUT * NLON_OUT;
    resample_kernel<<<(npix + 255) / 256, 256, 0, stream>>>(x, xr);
    disco_kernel<<<NLAT_OUT, 512, 0, stream>>>(xr, weight, psi_vals, psi_hi, psi_dw, out);
}